// DLTM_29386166239452
// MI455X (gfx1250) — compile-verified
//
#include <hip/hip_runtime.h>
#include <math.h>

typedef __attribute__((ext_vector_type(2))) float v2f;
typedef __attribute__((ext_vector_type(8))) float v8f;

#define F_TOTAL 2047
#define B_TOTAL 256
#define H_DIM   64
#define W_STRIDE 66   // padded: bank-conflict-free column reads, float2-aligned
#define E_STRIDE 66
#define MIN_STD 0.001f
#define MAX_STD 7.0f
#define TINY_F  1.1754944e-38f
#define HALF_LOG_2PI 0.91893853320467274178f

// LDS layout (float offsets)
#define W_OFF   0
#define E_OFF   (H_DIM * W_STRIDE)                 // 4224
#define M_OFF   (E_OFF + B_TOTAL * E_STRIDE)       // 21120
#define LOC_OFF (M_OFF + B_TOTAL)                  // 21376
#define IS_OFF  (LOC_OFF + H_DIM)                  // 21440
#define NLS_OFF (IS_OFF + H_DIM)                   // 21504
#define SMEM_FLOATS (NLS_OFF + H_DIM)              // 21568 floats = 86272 B

__device__ __forceinline__ float wave_max(float v) {
    #pragma unroll
    for (int off = 16; off > 0; off >>= 1) v = fmaxf(v, __shfl_xor(v, off, 32));
    return v;
}
__device__ __forceinline__ float wave_sum(float v) {
    #pragma unroll
    for (int off = 16; off > 0; off >>= 1) v += __shfl_xor(v, off, 32);
    return v;
}

// One workgroup processes one SPN node: 16 waves, each wave owns a 16-row
// batch tile of the 256x64 @ 64x64 mixture GEMM done with v_wmma_f32_16x16x4_f32.
__global__ __launch_bounds__(512, 1)
void spn_layer_kernel(const float* __restrict__ x,           // [B][F]
                      const float* __restrict__ sum_logits,  // [F][H][H]
                      const float* __restrict__ leaf_logits, // [F][2][H]
                      float* __restrict__ sum_ll,            // ws: [F][B][H]
                      float* __restrict__ out,               // [B][16]
                      int node_base)
{
    extern __shared__ float lds[];
    const int n    = node_base + (int)blockIdx.x;
    const int tid  = (int)threadIdx.x;
    const int wave = tid >> 5;        // 0..15
    const int lane = tid & 31;        // wave32

    // ---- Phase A: leaf params: loc, 1/scale, -log(scale)-0.5*log(2pi) ----
    if (tid < H_DIM) {
        float l0 = leaf_logits[(size_t)n * 2 * H_DIM + tid];
        float l1 = leaf_logits[(size_t)n * 2 * H_DIM + H_DIM + tid];
        float sig = 1.0f / (1.0f + __expf(-l1));
        float s = l1 * sig + 0.279f;                    // silu + 0.279
        s = fminf(fmaxf(s, MIN_STD), MAX_STD);
        lds[LOC_OFF + tid] = l0;
        lds[IS_OFF  + tid] = 1.0f / s;
        lds[NLS_OFF + tid] = -__logf(s) - HALF_LOG_2PI;
    }

    // ---- Phase B: W = softmax(sum_logits[n]) rows, stored row-major stride 66
    const float* SL = sum_logits + (size_t)n * H_DIM * H_DIM;
    #pragma unroll
    for (int i = wave; i < H_DIM; i += 16) {            // 4 rows per wave
        float a0 = SL[i * H_DIM + lane];
        float a1 = SL[i * H_DIM + lane + 32];
        float mx = wave_max(fmaxf(a0, a1));
        float e0 = __expf(a0 - mx), e1 = __expf(a1 - mx);
        float inv = 1.0f / wave_sum(e0 + e1);
        lds[W_OFF + i * W_STRIDE + lane]      = e0 * inv;
        lds[W_OFF + i * W_STRIDE + lane + 32] = e1 * inv;
    }
    __syncthreads();

    // ---- Phase C: prod = leaf_ll + children sum_ll; rowmax; exp -> E ----
    const bool has_kids = (2 * n + 1) < F_TOTAL;
    const float* c0 = sum_ll + (size_t)(2 * n + 1) * B_TOTAL * H_DIM;
    const float* c1 = sum_ll + (size_t)(2 * n + 2) * B_TOTAL * H_DIM;
    #pragma unroll 4
    for (int r = 0; r < 16; ++r) {
        int b = wave * 16 + r;
        float xv = x[(size_t)b * F_TOTAL + n];
        int j0 = lane, j1 = lane + 32;
        float cs0 = 0.0f, cs1 = 0.0f;
        if (has_kids) {
            cs0 = c0[b * H_DIM + j0] + c1[b * H_DIM + j0];
            cs1 = c0[b * H_DIM + j1] + c1[b * H_DIM + j1];
        }
        float z0 = (xv - lds[LOC_OFF + j0]) * lds[IS_OFF + j0];
        float z1 = (xv - lds[LOC_OFF + j1]) * lds[IS_OFF + j1];
        float p0 = fmaf(-0.5f * z0, z0, lds[NLS_OFF + j0]) + cs0;
        float p1 = fmaf(-0.5f * z1, z1, lds[NLS_OFF + j1]) + cs1;
        float mx = wave_max(fmaxf(p0, p1));
        lds[E_OFF + b * E_STRIDE + j0] = __expf(p0 - mx);
        lds[E_OFF + b * E_STRIDE + j1] = __expf(p1 - mx);
        if (lane == 0) lds[M_OFF + b] = mx;
    }
    __syncthreads();

    // ---- Phase D: mix = E(256x64) * W^T(64x64), f32 WMMA 16x16x4 ----
    // Wave w handles batch rows [16w, 16w+16), accumulating 4 N-tiles.
    const int m0   = wave * 16;
    const int mrow = m0 + (lane & 15);                  // A frag: M = lane%16
    const int kp   = (lane >> 4) * 2;                   // lanes16-31: K+2
    v8f acc[4];
    #pragma unroll
    for (int nt = 0; nt < 4; ++nt) acc[nt] = (v8f){};

    #pragma unroll
    for (int k0 = 0; k0 < H_DIM; k0 += 4) {
        v2f a;                                          // A: 16x4, 2 VGPRs
        a.x = lds[E_OFF + mrow * E_STRIDE + k0 + kp];
        a.y = lds[E_OFF + mrow * E_STRIDE + k0 + kp + 1];
        #pragma unroll
        for (int nt = 0; nt < 4; ++nt) {
            int col = nt * 16 + (lane & 15);            // B: 4x16 = W^T tile
            v2f bf;                                     // W^T[k][col] = W[col][k]
            bf.x = lds[W_OFF + col * W_STRIDE + k0 + kp];
            bf.y = lds[W_OFF + col * W_STRIDE + k0 + kp + 1];
            acc[nt] = __builtin_amdgcn_wmma_f32_16x16x4_f32(
                false, a, false, bf, (short)0, acc[nt], false, false);
        }
    }

    // ---- Phase E: sum_ll[n] = m + log(max(mix, TINY)); root -> out ----
    const int colb   = lane & 15;
    const int rowadd = (lane >> 4) * 8;                 // C layout: M=v / v+8
    #pragma unroll
    for (int nt = 0; nt < 4; ++nt) {
        #pragma unroll
        for (int v = 0; v < 8; ++v) {
            int row = m0 + v + rowadd;
            int col = nt * 16 + colb;
            float val = lds[M_OFF + row] + __logf(fmaxf(acc[nt][v], TINY_F));
            sum_ll[((size_t)n * B_TOTAL + row) * H_DIM + col] = val;
            if (n == 0 && col < 16) out[row * 16 + col] = val;
        }
    }
}

extern "C" void kernel_launch(void* const* d_in, const int* in_sizes, int n_in,
                              void* d_out, int out_size, void* d_ws, size_t ws_size,
                              hipStream_t stream) {
    (void)in_sizes; (void)n_in; (void)out_size; (void)ws_size;
    const float* x           = (const float*)d_in[0];
    const float* sum_logits  = (const float*)d_in[1];
    const float* leaf_logits = (const float*)d_in[2];
    float* out    = (float*)d_out;
    float* sum_ll = (float*)d_ws;   // [F][B][H] f32, fully overwritten bottom-up

    const size_t smem = SMEM_FLOATS * sizeof(float);
    // Tree levels bottom-up: height h nodes are [2^(10-h)-1 .. 2^(11-h)-2]
    for (int h = 0; h <= 10; ++h) {
        int count = 1 << (10 - h);
        int base  = count - 1;
        spn_layer_kernel<<<count, 512, smem, stream>>>(
            x, sum_logits, leaf_logits, sum_ll, out, base);
    }
}